// mmd_loss_30915174596665
// MI455X (gfx1250) — compile-verified
//
#include <hip/hip_runtime.h>
#include <math.h>

// ---------------------------------------------------------------------------
// MMD loss, fused:  G = T*T^T (fp32 WMMA, triangular), bandwidth analytic,
// 5-kernel Gaussian sum + signed reduction fused into the GEMM epilogue.
// ---------------------------------------------------------------------------

#define N_TOTAL 8192
#define B_HALF  4096
#define KDIM    1024
#define TILE    128
#define NT      (N_TOTAL / TILE)   // 64 -> 2080 triangular tiles
#define KC      32
#define LDK     36                  // padded LDS row stride (floats): 16B-aligned stores,
                                    // odd 128-bit stride (9 x 16B) -> conflict-free ds_load_b64

typedef __attribute__((ext_vector_type(2))) float v2f;
typedef __attribute__((ext_vector_type(4))) float v4f;
typedef __attribute__((ext_vector_type(8))) float v8f;

// workspace layout (float indices)
#define WS_LOSS    0
#define WS_SUMSQ   1
#define WS_NEGINV  4                    // 5 floats: -1/(bw*2^t)
#define WS_SQ      16                   // 8192 row squared-norms
#define WS_COLSUM  (16 + N_TOTAL)       // 1024 column sums
#define WS_ZERO_N  (WS_COLSUM + KDIM)   // 9232 floats to zero

__device__ __forceinline__ const float* row_ptr(const float* __restrict__ src,
                                                const float* __restrict__ tgt,
                                                int r) {
    return (r < B_HALF) ? (src + (size_t)r * KDIM)
                        : (tgt + (size_t)(r - B_HALF) * KDIM);
}

__global__ void init_ws_kernel(float* ws) {
    int i = blockIdx.x * blockDim.x + threadIdx.x;
    if (i < WS_ZERO_N) ws[i] = 0.0f;
}

// one block per row: sq[row] = sum_k T[row,k]^2 ; accumulate total into WS_SUMSQ
__global__ __launch_bounds__(256) void row_sq_kernel(const float* __restrict__ src,
                                                     const float* __restrict__ tgt,
                                                     float* ws) {
    const int row = blockIdx.x;
    const float* p = row_ptr(src, tgt, row);
    float s = 0.0f;
    for (int k = threadIdx.x; k < KDIM; k += 256) {
        float v = p[k];
        s += v * v;
    }
    __shared__ float red[256];
    red[threadIdx.x] = s;
    __syncthreads();
    for (int off = 128; off > 0; off >>= 1) {
        if (threadIdx.x < off) red[threadIdx.x] += red[threadIdx.x + off];
        __syncthreads();
    }
    if (threadIdx.x == 0) {
        ws[WS_SQ + row] = red[0];
        atomicAdd(&ws[WS_SUMSQ], red[0]);
    }
}

// colsum[k] = sum_r T[r,k]  (grid: x covers columns, y covers row chunks)
__global__ __launch_bounds__(256) void colsum_kernel(const float* __restrict__ src,
                                                     const float* __restrict__ tgt,
                                                     float* ws) {
    const int k  = blockIdx.x * 256 + threadIdx.x;
    const int r0 = blockIdx.y * 256;
    float s = 0.0f;
    for (int r = r0; r < r0 + 256; ++r) s += row_ptr(src, tgt, r)[k];
    atomicAdd(&ws[WS_COLSUM + k], s);
}

// bandwidth:  sum(L2) = 2n*sum(sq) - 2*||colsum||^2  (analytic, no L2 matrix)
__global__ __launch_bounds__(256) void bandwidth_kernel(float* ws) {
    __shared__ float red[256];
    float s = 0.0f;
    for (int k = threadIdx.x; k < KDIM; k += 256) {
        float c = ws[WS_COLSUM + k];
        s += c * c;
    }
    red[threadIdx.x] = s;
    __syncthreads();
    for (int off = 128; off > 0; off >>= 1) {
        if (threadIdx.x < off) red[threadIdx.x] += red[threadIdx.x + off];
        __syncthreads();
    }
    if (threadIdx.x == 0) {
        double n     = (double)N_TOTAL;
        double sumL2 = 2.0 * n * (double)ws[WS_SUMSQ] - 2.0 * (double)red[0];
        double bw    = sumL2 / (n * n - n) / 4.0;  // / KERNEL_MUL^(KERNEL_NUM//2)
        double m = 1.0;
        for (int t = 0; t < 5; ++t) {
            ws[WS_NEGINV + t] = (float)(-1.0 / (bw * m));
            m *= 2.0;
        }
    }
}

// Fused triangular GEMM + MMD epilogue.
// Block = 256 threads = 8 wave32. Each wave owns a 32x64 region (2x4 WMMA tiles).
__global__ __launch_bounds__(256) void mmd_gemm_kernel(const float* __restrict__ src,
                                                       const float* __restrict__ tgt,
                                                       float* ws) {
    // linear block index -> upper-triangular (ti, tj), tj >= ti
    int rem = blockIdx.x, ti = 0, rowlen = NT;
    while (rem >= rowlen) { rem -= rowlen; ++ti; --rowlen; }
    const int tj = ti + rem;
    const int i0 = ti * TILE;
    const int j0 = tj * TILE;

    __shared__ float As[TILE][LDK];
    __shared__ float Bs[TILE][LDK];

    const int tid   = threadIdx.x;
    const int lane  = tid & 31;
    const int wave  = tid >> 5;
    const int wi    = wave & 3;            // i strip: 4 x 32 rows
    const int wj    = wave >> 2;           // j strip: 2 x 64 rows
    const int lrow  = lane & 15;           // fragment row (M or N)
    const int khalf = (lane >> 4) << 1;    // lanes 16-31 hold K+2,K+3

    v8f acc[2][4] = {};

    for (int kk = 0; kk < KDIM; kk += KC) {
        // stage 128x32 of A-rows (i block) and B-rows (j block) into LDS
        #pragma unroll
        for (int l = 0; l < 4; ++l) {
            const int fi = tid + 256 * l;   // float4 index, 0..1023
            const int r  = fi >> 3;         // 8 float4 per 32-float row
            const int c  = (fi & 7) << 2;
            const float* pa = row_ptr(src, tgt, i0 + r);
            const float* pb = row_ptr(src, tgt, j0 + r);
            *(v4f*)&As[r][c] = *(const v4f*)(pa + kk + c);
            *(v4f*)&Bs[r][c] = *(const v4f*)(pb + kk + c);
        }
        __syncthreads();

        #pragma unroll
        for (int k2 = 0; k2 < KC; k2 += 4) {
            v2f afrag[2], bfrag[4];
            #pragma unroll
            for (int a = 0; a < 2; ++a)
                afrag[a] = *(const v2f*)&As[wi * 32 + a * 16 + lrow][k2 + khalf];
            #pragma unroll
            for (int b = 0; b < 4; ++b)
                bfrag[b] = *(const v2f*)&Bs[wj * 64 + b * 16 + lrow][k2 + khalf];
            #pragma unroll
            for (int a = 0; a < 2; ++a)
                #pragma unroll
                for (int b = 0; b < 4; ++b)
                    acc[a][b] = __builtin_amdgcn_wmma_f32_16x16x4_f32(
                        /*neg_a=*/false, afrag[a], /*neg_b=*/false, bfrag[b],
                        /*c_mod=*/(short)0, acc[a][b],
                        /*reuse_a=*/false, /*reuse_b=*/false);
        }
        __syncthreads();
    }

    // ---- fused epilogue: L2 -> 5 exps -> signed sum -----------------------
    float nib[5];
    #pragma unroll
    for (int t = 0; t < 5; ++t) nib[t] = ws[WS_NEGINV + t];
    const float* sq = ws + WS_SQ;

    const float wgt   = (ti == tj) ? 1.0f : 2.0f;   // symmetry weight
    const int   nn    = lane & 15;                  // N index in C tile
    const int   mbase = (lane >> 4) * 8;            // lanes 16-31: M = v + 8

    float lsum = 0.0f;
    #pragma unroll
    for (int a = 0; a < 2; ++a) {
        #pragma unroll
        for (int b = 0; b < 4; ++b) {
            const int   j   = j0 + wj * 64 + b * 16 + nn;
            const float sqj = sq[j];
            const float sj  = (j < B_HALF) ? 1.0f : -1.0f;
            #pragma unroll
            for (int v = 0; v < 8; ++v) {
                const int   i  = i0 + wi * 32 + a * 16 + mbase + v;
                const float g  = acc[a][b][v];
                const float l2 = sq[i] + sqj - 2.0f * g;
                const float si = (i < B_HALF) ? 1.0f : -1.0f;
                float ksum = 0.0f;
                #pragma unroll
                for (int t = 0; t < 5; ++t) ksum += __expf(l2 * nib[t]);
                lsum += si * sj * ksum;
            }
        }
    }
    lsum *= wgt;

    __shared__ float red[256];
    red[tid] = lsum;
    __syncthreads();
    for (int off = 128; off > 0; off >>= 1) {
        if (tid < off) red[tid] += red[tid + off];
        __syncthreads();
    }
    if (tid == 0) atomicAdd(&ws[WS_LOSS], red[0]);
}

__global__ void finalize_kernel(const float* ws, float* out) {
    out[0] = ws[WS_LOSS] / ((float)B_HALF * (float)B_HALF);
}

extern "C" void kernel_launch(void* const* d_in, const int* in_sizes, int n_in,
                              void* d_out, int out_size, void* d_ws, size_t ws_size,
                              hipStream_t stream) {
    const float* src = (const float*)d_in[0];
    const float* tgt = (const float*)d_in[1];
    float* ws  = (float*)d_ws;
    float* out = (float*)d_out;

    init_ws_kernel<<<(WS_ZERO_N + 255) / 256, 256, 0, stream>>>(ws);
    row_sq_kernel<<<N_TOTAL, 256, 0, stream>>>(src, tgt, ws);
    colsum_kernel<<<dim3(KDIM / 256, N_TOTAL / 256), 256, 0, stream>>>(src, tgt, ws);
    bandwidth_kernel<<<1, 256, 0, stream>>>(ws);
    mmd_gemm_kernel<<<NT * (NT + 1) / 2, 256, 0, stream>>>(src, tgt, ws);
    finalize_kernel<<<1, 1, 0, stream>>>(ws, out);
}